// ResidualBottleneck_51634096832599
// MI455X (gfx1250) — compile-verified
//
#include <hip/hip_runtime.h>

// Problem constants (from the reference)
#define BDIM   8
#define TDIM   4096
#define BT     (BDIM * TDIM)   // 32768 tokens
#define HID    512
#define DIM    256
#define KCODES 1024
#define NQ     8

typedef __attribute__((ext_vector_type(2))) float v2f;
typedef __attribute__((ext_vector_type(8))) float v8f;

// FP32 WMMA: D(16x16,f32) = A(16x4,f32) * B(4x16,f32) + C
// A frag: lane m = lane&15; vgpr0 = K {0|2} (koff by lane>=16), vgpr1 = K {1|3}
// B frag: lane n = lane&15; same K split -> per-lane contiguous float2 at (n, k)
// C frag: vgpr i -> row (i + 8*(lane>>4)), col = lane&15
__device__ __forceinline__ v8f wmma_f32(v2f a, v2f b, v8f c) {
  return __builtin_amdgcn_wmma_f32_16x16x4_f32(false, a, false, b, (short)0, c,
                                               false, false);
}

#define XPAD 4
#define RPAD 4

// ---------------------------------------------------------------------------
// Kernel 1: h = x @ W_in^T + b_in ; resid = h ; zero the commitment scalar.
// grid = BT/16 blocks of 512 threads (16 waves). Each wave owns one 16-wide
// N-tile of DIM=256. x tile (16x512) staged in LDS with padded stride.
// ---------------------------------------------------------------------------
__global__ __launch_bounds__(512)
void proj_in_kernel(const float* __restrict__ x, const float* __restrict__ Wi,
                    const float* __restrict__ bi, float* __restrict__ h,
                    float* __restrict__ resid, float* __restrict__ com) {
  __shared__ float xs[16 * (HID + XPAD)];
  const int tid = threadIdx.x;
  const int m0 = blockIdx.x * 16;

  // 16 full rows of x are contiguous: vectorized cooperative load.
  const float4* xsrc = (const float4*)(x + (size_t)m0 * HID);
  for (int f = tid; f < 16 * HID / 4; f += 512) {
    float4 v = xsrc[f];
    int e = f * 4, row = e >> 9, colc = e & (HID - 1);
    float* dst = &xs[row * (HID + XPAD) + colc];
    dst[0] = v.x; dst[1] = v.y; dst[2] = v.z; dst[3] = v.w;
  }
  __syncthreads();

  const int wave = tid >> 5, lane = tid & 31;
  const int l15 = lane & 15;
  const int koff = (lane >> 4) * 2;
  const int n0 = wave * 16;

  const float* arow = &xs[l15 * (HID + XPAD) + koff];
  const float* brow = Wi + (size_t)(n0 + l15) * HID + koff;   // B(k,n)=W_in[n,k]

  v8f acc = {};
  #pragma unroll 8
  for (int k = 0; k < HID; k += 4) {
    v2f a = *(const v2f*)(arow + k);
    v2f b = *(const v2f*)(brow + k);
    acc = wmma_f32(a, b, acc);
  }

  const float bn = bi[n0 + l15];
  const int mh = (lane >> 4) * 8;
  #pragma unroll
  for (int i = 0; i < 8; ++i) {
    size_t g = (size_t)(m0 + i + mh) * DIM + n0 + l15;
    float hv = acc[i] + bn;
    h[g] = hv;
    resid[g] = hv;
  }
  if (blockIdx.x == 0 && tid == 0) *com = 0.0f;
}

// ---------------------------------------------------------------------------
// Kernel 2: per-code squared norms ||c_k||^2, one wave per code.
// ---------------------------------------------------------------------------
__global__ __launch_bounds__(256)
void cnorm_kernel(const float* __restrict__ cb, float* __restrict__ cn) {
  const int code = (blockIdx.x * 256 + threadIdx.x) >> 5;   // 0 .. NQ*K-1
  const int lane = threadIdx.x & 31;
  const float* row = cb + (size_t)code * DIM;
  float s = 0.f;
  #pragma unroll
  for (int j = 0; j < DIM; j += 32) { float v = row[j + lane]; s += v * v; }
  #pragma unroll
  for (int off = 16; off > 0; off >>= 1) s += __shfl_xor(s, off, 32);
  if (lane == 0) cn[code] = s;
}

// ---------------------------------------------------------------------------
// Kernel 3 (launched NQ times): one RVQ step.
// Block = 16 rows; 16 waves x 4 tiles cover all 1024 codes via WMMA distance
// scores s = ||c||^2 - 2*r.c  (||r||^2 constant per row -> dropped for argmin).
// Then gather nearest code, update residual, emit q1/q2, commitment partial.
// ---------------------------------------------------------------------------
__global__ __launch_bounds__(512)
void vq_step_kernel(float* __restrict__ resid, const float* __restrict__ cb,
                    const float* __restrict__ cn, float* __restrict__ q1,
                    float* __restrict__ q2, float* __restrict__ com, int iq) {
  __shared__ float rs[16 * (DIM + RPAD)];
  __shared__ float redv[16][16];
  __shared__ int   redi[16][16];
  __shared__ int   idxf[16];
  __shared__ float csum_sh;

  const int tid = threadIdx.x;
  const int m0 = blockIdx.x * 16;

  const float4* rsrc = (const float4*)(resid + (size_t)m0 * DIM);
  for (int f = tid; f < 16 * DIM / 4; f += 512) {
    float4 v = rsrc[f];
    int e = f * 4, row = e >> 8, colc = e & (DIM - 1);
    float* dst = &rs[row * (DIM + RPAD) + colc];
    dst[0] = v.x; dst[1] = v.y; dst[2] = v.z; dst[3] = v.w;
  }
  if (tid == 0) csum_sh = 0.f;
  __syncthreads();

  const int wave = tid >> 5, lane = tid & 31;
  const int l15 = lane & 15;
  const int koff = (lane >> 4) * 2;
  const float* arow = &rs[l15 * (DIM + RPAD) + koff];

  float bestv[8];
  int   besti[8];
  #pragma unroll
  for (int i = 0; i < 8; ++i) { bestv[i] = 3.4e38f; besti[i] = 0; }

  for (int t = 0; t < 4; ++t) {
    const int n0 = (wave + 16 * t) * 16;
    const float* brow = cb + (size_t)(n0 + l15) * DIM + koff;
    v8f acc = {};
    #pragma unroll 8
    for (int k = 0; k < DIM; k += 4) {
      v2f a = *(const v2f*)(arow + k);
      v2f b = *(const v2f*)(brow + k);
      acc = wmma_f32(a, b, acc);
    }
    const float cnv = cn[n0 + l15];
    #pragma unroll
    for (int i = 0; i < 8; ++i) {
      float s = cnv - 2.0f * acc[i];   // row = i + 8*(lane>>4), col = l15
      int id = n0 + l15;
      // 16-lane butterfly min (stays inside each 16-lane half), low-idx ties
      #pragma unroll
      for (int off = 1; off < 16; off <<= 1) {
        float ov = __shfl_xor(s, off, 32);
        int   oi = __shfl_xor(id, off, 32);
        if (ov < s || (ov == s && oi < id)) { s = ov; id = oi; }
      }
      if (s < bestv[i] || (s == bestv[i] && id < besti[i])) {
        bestv[i] = s; besti[i] = id;
      }
    }
  }

  if (l15 == 0) {                       // lanes 0 (rows 0..7) and 16 (rows 8..15)
    const int rbase = (lane >> 4) * 8;
    #pragma unroll
    for (int i = 0; i < 8; ++i) {
      redv[wave][rbase + i] = bestv[i];
      redi[wave][rbase + i] = besti[i];
    }
  }
  __syncthreads();

  if (tid < 16) {
    float bv = redv[0][tid];
    int   bi2 = redi[0][tid];
    for (int w = 1; w < 16; ++w) {
      float v = redv[w][tid]; int ii = redi[w][tid];
      if (v < bv || (v == bv && ii < bi2)) { bv = v; bi2 = ii; }
    }
    idxf[tid] = bi2;
  }
  __syncthreads();

  // Gather + residual update + commitment partial (4096 elems / 512 threads)
  float csum = 0.f;
  #pragma unroll
  for (int j = 0; j < 8; ++j) {
    int e = tid + 512 * j;
    int row = e >> 8, colc = e & (DIM - 1);
    int id = idxf[row];
    float q = cb[(size_t)id * DIM + colc];
    float r = rs[row * (DIM + RPAD) + colc];
    float d = q - r;                    // (stop_grad(q) - residual)
    csum += d * d;
    size_t g = (size_t)(m0 + row) * DIM + colc;
    resid[g] = r - q;                   // residual - q_st  (q_st == q in value)
    if (iq == 0)      q1[g] = q;
    else if (iq == 1) q2[g] = q;
  }
  #pragma unroll
  for (int off = 16; off > 0; off >>= 1) csum += __shfl_xor(csum, off, 32);
  if (lane == 0) atomicAdd(&csum_sh, csum);
  __syncthreads();
  if (tid == 0)
    atomicAdd(com, csum_sh * (1.0f / ((float)BT * (float)DIM * (float)NQ)));
}

// ---------------------------------------------------------------------------
// Kernel 4: out = (h - resid_final) @ W_out^T + b_out
// (sum of quantized layers == h - final residual; avoids a qsum RMW buffer)
// ---------------------------------------------------------------------------
__global__ __launch_bounds__(512)
void proj_out_kernel(const float* __restrict__ h, const float* __restrict__ resid,
                     const float* __restrict__ Wo, const float* __restrict__ bo,
                     float* __restrict__ out) {
  __shared__ float qs[16 * (DIM + RPAD)];
  const int tid = threadIdx.x;
  const int m0 = blockIdx.x * 16;

  const float4* hsrc = (const float4*)(h + (size_t)m0 * DIM);
  const float4* rsrc = (const float4*)(resid + (size_t)m0 * DIM);
  for (int f = tid; f < 16 * DIM / 4; f += 512) {
    float4 hv = hsrc[f];
    float4 rv = rsrc[f];
    int e = f * 4, row = e >> 8, colc = e & (DIM - 1);
    float* dst = &qs[row * (DIM + RPAD) + colc];
    dst[0] = hv.x - rv.x; dst[1] = hv.y - rv.y;
    dst[2] = hv.z - rv.z; dst[3] = hv.w - rv.w;
  }
  __syncthreads();

  const int wave = tid >> 5, lane = tid & 31;
  const int l15 = lane & 15;
  const int koff = (lane >> 4) * 2;
  const float* arow = &qs[l15 * (DIM + RPAD) + koff];
  const int mh = (lane >> 4) * 8;

  for (int t = 0; t < 2; ++t) {                  // HID=512 -> 32 N-tiles / 16 waves
    const int n0 = (wave + 16 * t) * 16;
    const float* brow = Wo + (size_t)(n0 + l15) * DIM + koff;  // B(k,n)=W_out[n,k]
    v8f acc = {};
    #pragma unroll 8
    for (int k = 0; k < DIM; k += 4) {
      v2f a = *(const v2f*)(arow + k);
      v2f b = *(const v2f*)(brow + k);
      acc = wmma_f32(a, b, acc);
    }
    const float bn = bo[n0 + l15];
    #pragma unroll
    for (int i = 0; i < 8; ++i)
      out[(size_t)(m0 + i + mh) * HID + n0 + l15] = acc[i] + bn;
  }
}

// ---------------------------------------------------------------------------
extern "C" void kernel_launch(void* const* d_in, const int* in_sizes, int n_in,
                              void* d_out, int out_size, void* d_ws, size_t ws_size,
                              hipStream_t stream) {
  (void)in_sizes; (void)n_in; (void)out_size; (void)ws_size;
  const float* x   = (const float*)d_in[0];   // [B,T,HID]
  const float* Wi  = (const float*)d_in[1];   // [DIM,HID]
  const float* bi  = (const float*)d_in[2];   // [DIM]
  const float* Wo  = (const float*)d_in[3];   // [HID,DIM]
  const float* bo  = (const float*)d_in[4];   // [HID]
  const float* cbs = (const float*)d_in[5];   // [NQ,K,DIM]

  float* out = (float*)d_out;                  // [BT,HID]
  float* q1  = out + (size_t)BT * HID;         // [BT,DIM]
  float* q2  = q1 + (size_t)BT * DIM;          // [BT,DIM]
  float* com = q2 + (size_t)BT * DIM;          // scalar

  float* h     = (float*)d_ws;                 // BT*DIM
  float* resid = h + (size_t)BT * DIM;         // BT*DIM
  float* cnorm = resid + (size_t)BT * DIM;     // NQ*K

  proj_in_kernel<<<BT / 16, 512, 0, stream>>>(x, Wi, bi, h, resid, com);
  cnorm_kernel<<<(NQ * KCODES) / 8, 256, 0, stream>>>(cbs, cnorm);
  for (int iq = 0; iq < NQ; ++iq) {
    vq_step_kernel<<<BT / 16, 512, 0, stream>>>(
        resid, cbs + (size_t)iq * KCODES * DIM, cnorm + (size_t)iq * KCODES,
        q1, q2, com, iq);
  }
  proj_out_kernel<<<BT / 16, 512, 0, stream>>>(h, resid, Wo, bo, out);
}